// SDFs2Grid_18915035972381
// MI455X (gfx1250) — compile-verified
//
#include <hip/hip_runtime.h>

typedef __attribute__((ext_vector_type(2))) float v2f;
typedef __attribute__((ext_vector_type(8))) float v8f;

#define GXD 96
#define GXYZ (96 * 96 * 96)
#define BIG_VAL 1.0e10f

// One wave of 32 lanes handles 16 grid points for BOTH batches:
//   lane l: point slot = l&15, batch = l>>4, instances (batch*8 .. batch*8+7)
// Three V_WMMA_F32_16X16X4_F32 ops compute g = A*[p;1] for all 16 instances
// x 16 points; D-layout puts instance r of the lane's batch in acc VGPR r.
__global__ __launch_bounds__(256) void sdf2grid_wmma_kernel(
    const float* __restrict__ sdfs,        // (M, sdf_stride)
    const int*   __restrict__ sdf_shapes,  // (M, 3)
    const int*   __restrict__ indices,     // (2, 8)
    const float* __restrict__ sdf_poses,   // (2, 8, 7)  trans + quat
    const float* __restrict__ sdf_widths,  // (2, 8, 3)
    float*       __restrict__ out,         // (2, 96, 96, 96)
    int sdf_stride)
{
    const int lane = threadIdx.x & 31;
    const int half = lane >> 4;        // == batch id for the gather phase
    const int slot = lane & 15;        // point slot / A-matrix row (instance id)
    const int gwave = (blockIdx.x * blockDim.x + threadIdx.x) >> 5;
    const int pt = gwave * 16 + slot;  // 0 .. 884735  (884736 % 16 == 0)

    // ---------- grid point (B operand: column pt%16 = [px,py,pz,1]) ----------
    const int iz = pt % GXD;
    const int tt = pt / GXD;
    const int iy = tt % GXD;
    const int ix = tt / GXD;
    const float px = -1.2f + ((float)ix + 0.5f) * 0.025f;
    const float py = -1.2f + ((float)iy + 0.5f) * 0.025f;
    const float pz = -1.2f + ((float)iz + 0.5f) * 0.025f;

    // B[k][n]: lane supplies k = 2*half + {0,1} at column n = slot
    v2f vb;
    vb.x = half ? pz : px;     // k = 0 (px) or k = 2 (pz)
    vb.y = half ? 1.0f : py;   // k = 1 (py) or k = 3 (1.0)

    // ---------- instance affine (A operand: row = slot = instance) ----------
    // g = Diag(1/w) * R(q)^T * (p - t) - 0.5  ==>  g[c] = A[c][0..2].p + A[c][3]
    const float* P = sdf_poses + slot * 7;
    float tx = P[0], ty = P[1], tz = P[2];
    float qx = P[3], qy = P[4], qz = P[5], qw = P[6];
    const float rn = rsqrtf(qx * qx + qy * qy + qz * qz + qw * qw);
    qx *= rn; qy *= rn; qz *= rn; qw *= rn;
    const float R00 = 1.f - 2.f * (qy * qy + qz * qz);
    const float R01 = 2.f * (qx * qy - qz * qw);
    const float R02 = 2.f * (qx * qz + qy * qw);
    const float R10 = 2.f * (qx * qy + qz * qw);
    const float R11 = 1.f - 2.f * (qx * qx + qz * qz);
    const float R12 = 2.f * (qy * qz - qx * qw);
    const float R20 = 2.f * (qx * qz - qy * qw);
    const float R21 = 2.f * (qy * qz + qx * qw);
    const float R22 = 1.f - 2.f * (qx * qx + qy * qy);
    const float* Wd = sdf_widths + slot * 3;
    float A[3][4];
    {
        const float iw0 = 1.f / Wd[0], iw1 = 1.f / Wd[1], iw2 = 1.f / Wd[2];
        A[0][0] = R00 * iw0; A[0][1] = R10 * iw0; A[0][2] = R20 * iw0;  // (R^T row 0)/w0
        A[1][0] = R01 * iw1; A[1][1] = R11 * iw1; A[1][2] = R21 * iw1;
        A[2][0] = R02 * iw2; A[2][1] = R12 * iw2; A[2][2] = R22 * iw2;
        A[0][3] = -(A[0][0] * tx + A[0][1] * ty + A[0][2] * tz) - 0.5f;
        A[1][3] = -(A[1][0] * tx + A[1][1] * ty + A[1][2] * tz) - 0.5f;
        A[2][3] = -(A[2][0] * tx + A[2][1] * ty + A[2][2] * tz) - 0.5f;
    }
    const int k0 = half * 2;   // lane supplies A[row][k0], A[row][k0+1]
    v2f a0, a1, a2;
    a0.x = A[0][k0]; a0.y = A[0][k0 + 1];
    a1.x = A[1][k0]; a1.y = A[1][k0 + 1];
    a2.x = A[2][k0]; a2.y = A[2][k0 + 1];

    // ---------- 3 WMMAs: D[c][inst][point] = g coordinate c ----------
    const v8f zero = {};
    v8f gxv = __builtin_amdgcn_wmma_f32_16x16x4_f32(false, a0, false, vb,
                                                    (short)0, zero, false, false);
    v8f gyv = __builtin_amdgcn_wmma_f32_16x16x4_f32(false, a1, false, vb,
                                                    (short)0, zero, false, false);
    v8f gzv = __builtin_amdgcn_wmma_f32_16x16x4_f32(false, a2, false, vb,
                                                    (short)0, zero, false, false);

    // ---------- trilinear + inside mask + min over this batch's 8 ----------
    const int b = half;
    float m = BIG_VAL;
#pragma unroll
    for (int r = 0; r < 8; ++r) {
        const float gx = gxv[r];
        const float gy = gyv[r];
        const float gz = gzv[r];
        const int sidx = indices[b * 8 + r];
        const int SXs = sdf_shapes[sidx * 3 + 0];
        const int SYs = sdf_shapes[sidx * 3 + 1];
        const int SZs = sdf_shapes[sidx * 3 + 2];
        const float ex = (float)(SXs - 1);
        const float ey = (float)(SYs - 1);
        const float ez = (float)(SZs - 1);
        const bool inside = (gx >= 0.f) && (gx <= ex) && (gy >= 0.f) &&
                            (gy <= ey) && (gz >= 0.f) && (gz <= ez);
        const float fx = fminf(fmaxf(gx, 0.f), ex);
        const float fy = fminf(fmaxf(gy, 0.f), ey);
        const float fz = fminf(fmaxf(gz, 0.f), ez);
        const int x0 = (int)fx, y0 = (int)fy, z0 = (int)fz;   // floor (f >= 0)
        const float wx = fx - (float)x0;
        const float wy = fy - (float)y0;
        const float wz = fz - (float)z0;
        const int x1 = min(x0 + 1, SXs - 1);
        const int y1 = min(y0 + 1, SYs - 1);
        const int z1 = min(z0 + 1, SZs - 1);
        const float* v = sdfs + (size_t)sidx * (size_t)sdf_stride;
        const int rx0 = x0 * SYs, rx1 = x1 * SYs;
        const int p00 = (rx0 + y0) * SZs, p01 = (rx0 + y1) * SZs;
        const int p10 = (rx1 + y0) * SZs, p11 = (rx1 + y1) * SZs;
        const float c000 = v[p00 + z0], c001 = v[p00 + z1];
        const float c010 = v[p01 + z0], c011 = v[p01 + z1];
        const float c100 = v[p10 + z0], c101 = v[p10 + z1];
        const float c110 = v[p11 + z0], c111 = v[p11 + z1];
        const float c00 = c000 + wz * (c001 - c000);
        const float c01 = c010 + wz * (c011 - c010);
        const float c10 = c100 + wz * (c101 - c100);
        const float c11 = c110 + wz * (c111 - c110);
        const float c0 = c00 + wy * (c01 - c00);
        const float c1 = c10 + wy * (c11 - c10);
        const float acc = c0 + wx * (c1 - c0);
        m = fminf(m, inside ? acc : BIG_VAL);
    }
    out[(size_t)b * GXYZ + pt] = m;
}

extern "C" void kernel_launch(void* const* d_in, const int* in_sizes, int n_in,
                              void* d_out, int out_size, void* d_ws, size_t ws_size,
                              hipStream_t stream) {
    (void)n_in; (void)out_size; (void)d_ws; (void)ws_size;
    const float* sdfs    = (const float*)d_in[0];
    const int*   shapes  = (const int*)d_in[1];
    const int*   indices = (const int*)d_in[2];
    const float* poses   = (const float*)d_in[3];
    const float* widths  = (const float*)d_in[4];
    float*       out     = (float*)d_out;

    const int M = in_sizes[1] / 3;                 // number of stored SDFs
    const int sdf_stride = in_sizes[0] / (M > 0 ? M : 1);

    // 884736 points / 16 points-per-wave = 55296 waves; 8 waves/block -> 6912 blocks
    const int threads = 256;
    const int blocks = (GXYZ / 16) * 32 / threads; // 6912
    sdf2grid_wmma_kernel<<<blocks, threads, 0, stream>>>(
        sdfs, shapes, indices, poses, widths, out, sdf_stride);
}